// SoftRasterizer_43087111914141
// MI455X (gfx1250) — compile-verified
//
#include <hip/hip_runtime.h>
#include <stdint.h>

// ---------------- problem constants ----------------
#define IMGS   128
#define PIX    (IMGS*IMGS)          // 16384
#define NFACE  10000
#define F2     20000                // faces + reversed copies
#define F2P    20480                // padded to NSPLIT*NTILES*TILE
#define NSPLIT 10                   // face-dimension splits (parallelism)
#define FPS    (F2P/NSPLIT)         // 2048 faces per split
#define TILE   128                  // faces per LDS tile
#define NTILES (FPS/TILE)           // 16
#define RECF   32                   // floats per face record
#define RECB   128                  // bytes per face record

#define SIGMA_INV 1e6f
#define INV_G     1e6f
#define THR       1e-10f            // DIST_EPS*SIGMA
#define BG_EPS    1e-3f

// workspace layout (floats)
#define WS_PART_OFF ((size_t)F2P*RECF)      // 655360 floats
#define PART_STRIDE ((size_t)NSPLIT*PIX)    // per-component stride

// ---------------- CDNA5 async global->LDS helpers ----------------
__device__ __forceinline__ unsigned lds_off(const void* p) {
  // generic (flat) address of LDS: low 32 bits are the wave-relative LDS byte offset
  return (unsigned)(unsigned long long)p;
}

__device__ __forceinline__ void async_b128(unsigned lds, const char* sbase, unsigned goff) {
  // GLOBAL_LOAD_ASYNC_TO_LDS_B128 (GVS mode): copies 16 bytes from saddr + vaddr into LDS at offset vdst
  asm volatile("global_load_async_to_lds_b128 %0, %1, %2"
               :: "v"(lds), "v"(goff), "s"(sbase)
               : "memory");
}

__device__ __forceinline__ void wait_async_le4() {
  asm volatile("s_wait_asynccnt 4" ::: "memory");
}
__device__ __forceinline__ void wait_async_le0() {
  asm volatile("s_wait_asynccnt 0" ::: "memory");
}

// ---------------- kernel 1: per-face precomputation ----------------
// record layout (float index):
//  0:A0x 1:A0y 2:A1x 3:A1y | 4:x2 5:y2 6:x0 7:y0 | 8:x1 9:y1 10:e0x 11:e0y
// 12:il0 13:e1x 14:e1y 15:il1 | 16:e2x 17:e2y 18:il2 19:rz0
// 20:rz1 21:rz2 22:flags(0=invalid,1=back,2=front) 23:t00
// 24:t01 25:t02 26:t10 27:t11 | 28:t12 29:t20 30:t21 31:t22
__global__ __launch_bounds__(256) void sr_setup(
    const float* __restrict__ verts, const int* __restrict__ faces,
    const float* __restrict__ attr, float* __restrict__ rec) {
  int j = blockIdx.x * 256 + threadIdx.x;
  if (j >= F2P) return;
  float* r = rec + (size_t)j * RECF;
  if (j >= F2) {
#pragma unroll
    for (int k = 0; k < RECF; ++k) r[k] = 0.0f;
    return;
  }
  bool rev = (j >= NFACE);
  int fi = rev ? j - NFACE : j;
  int i0 = faces[fi*3+0], i1 = faces[fi*3+1], i2 = faces[fi*3+2];
  if (rev) { int t = i0; i0 = i2; i2 = t; }   // vertex order reversed

  float x0 = -verts[i0*3+0], y0 = verts[i0*3+1], z0 = verts[i0*3+2];
  float x1 = -verts[i1*3+0], y1 = verts[i1*3+1], z1 = verts[i1*3+2];
  float x2 = -verts[i2*3+0], y2 = verts[i2*3+1], z2 = verts[i2*3+2];

  float det = (y1 - y2)*(x0 - x2) + (x2 - x1)*(y0 - y2);
  if (fabsf(det) < 1e-10f) det = (det < 0.0f) ? -1e-10f : 1e-10f;
  float idet = 1.0f / det;

  float e0x = x1 - x0, e0y = y1 - y0;
  float e1x = x2 - x1, e1y = y2 - y1;
  float e2x = x0 - x2, e2y = y0 - y2;
  float il0 = 1.0f / fmaxf(e0x*e0x + e0y*e0y, 1e-12f);
  float il1 = 1.0f / fmaxf(e1x*e1x + e1y*e1y, 1e-12f);
  float il2 = 1.0f / fmaxf(e2x*e2x + e2y*e2y, 1e-12f);

  bool front = (y2 - y0)*(x1 - x0) < (y1 - y0)*(x2 - x0);

  const float* A = attr + (size_t)fi * 9;
  float t00,t01,t02,t10,t11,t12,t20,t21,t22;
  if (!rev) {
    t00=A[0]; t01=A[1]; t02=A[2]; t10=A[3]; t11=A[4]; t12=A[5]; t20=A[6]; t21=A[7]; t22=A[8];
  } else {
    t00=A[6]; t01=A[7]; t02=A[8]; t10=A[3]; t11=A[4]; t12=A[5]; t20=A[0]; t21=A[1]; t22=A[2];
  }

  r[0]=(y1-y2)*idet; r[1]=(x2-x1)*idet; r[2]=(y2-y0)*idet; r[3]=(x0-x2)*idet;
  r[4]=x2;  r[5]=y2;  r[6]=x0;  r[7]=y0;
  r[8]=x1;  r[9]=y1;  r[10]=e0x; r[11]=e0y;
  r[12]=il0; r[13]=e1x; r[14]=e1y; r[15]=il1;
  r[16]=e2x; r[17]=e2y; r[18]=il2; r[19]=1.0f/z0;
  r[20]=1.0f/z1; r[21]=1.0f/z2; r[22]= front ? 2.0f : 1.0f; r[23]=t00;
  r[24]=t01; r[25]=t02; r[26]=t10; r[27]=t11;
  r[28]=t12; r[29]=t20; r[30]=t21; r[31]=t22;
}

// ---------------- per-face rasterization body (LDS-broadcast record) ----------------
struct PxState { float m, s, cr, cg, cb, prod; };

__device__ __forceinline__ void raster_face(const float4* __restrict__ fr,
                                            float px, float py, PxState& st) {
  float4 q0 = fr[0], q1 = fr[1], q2 = fr[2], q3 = fr[3], q4 = fr[4];

  // barycentrics
  float dx2 = px - q1.x, dy2 = py - q1.y;
  float w0 = q0.x*dx2 + q0.y*dy2;
  float w1 = q0.z*dx2 + q0.w*dy2;
  float w2 = 1.0f - w0 - w1;
  bool inside = (w0 >= 0.0f) && (w1 >= 0.0f) && (w2 >= 0.0f);

  // squared distances to the 3 edges
  float dx0 = px - q1.z, dy0 = py - q1.w;
  float tt = fminf(fmaxf((dx0*q2.z + dy0*q2.w) * q3.x, 0.0f), 1.0f);
  float rx = dx0 - tt*q2.z, ry = dy0 - tt*q2.w;
  float dA = rx*rx + ry*ry;
  float dx1 = px - q2.x, dy1 = py - q2.y;
  tt = fminf(fmaxf((dx1*q3.y + dy1*q3.z) * q3.w, 0.0f), 1.0f);
  rx = dx1 - tt*q3.y; ry = dy1 - tt*q3.z;
  float dB = rx*rx + ry*ry;
  tt = fminf(fmaxf((dx2*q4.x + dy2*q4.y) * q4.z, 0.0f), 1.0f);
  rx = dx2 - tt*q4.x; ry = dy2 - tt*q4.y;
  float dC = rx*rx + ry*ry;
  float dis = fminf(fminf(dA, dB), dC);

  if (inside || dis < THR) {
    float4 q5 = fr[5];
    if (q5.z != 0.0f) {                              // valid face
      float sgn = inside ? 1.0f : -1.0f;
      float Dp = 1.0f / (1.0f + __expf(-sgn * dis * SIGMA_INV));
      st.prod *= (1.0f - Dp);

      float c0 = fminf(fmaxf(w0, 0.0f), 1.0f);
      float c1 = fminf(fmaxf(w1, 0.0f), 1.0f);
      float c2 = fminf(fmaxf(w2, 0.0f), 1.0f);
      float inv = 1.0f / fmaxf(c0 + c1 + c2, 1e-5f);
      c0 *= inv; c1 *= inv; c2 *= inv;
      float zpi = c0*q4.w + c1*q5.x + c2*q5.y;
      if (fabsf(zpi) < 1e-12f) zpi = 1e-12f;
      float zp = 1.0f / zpi;

      if (q5.z > 1.5f && zp >= 1.0f && zp <= 100.0f) {   // front && depth in range
        float zpn = (100.0f - zp) * (1.0f/99.0f);
        if (zpn > st.m) {                                 // online softmax rescale
          float sc = __expf((st.m - zpn) * INV_G);
          st.s *= sc; st.cr *= sc; st.cg *= sc; st.cb *= sc; st.m = zpn;
        }
        float e = __expf((zpn - st.m) * INV_G) * Dp;
        float4 q6 = fr[6], q7 = fr[7];
        st.s  += e;
        st.cr += e * (c0*q5.w + c1*q6.z + c2*q7.y);
        st.cg += e * (c0*q6.x + c1*q6.w + c2*q7.z);
        st.cb += e * (c0*q6.y + c1*q7.x + c2*q7.w);
      }
    }
  }
}

// ---------------- kernel 2: rasterize one face-split over a 16x16 pixel tile ----------------
__global__ __launch_bounds__(256) void sr_render(
    const float* __restrict__ rec, float* __restrict__ part) {
  __shared__ __align__(16) float tile[2][TILE * RECF];   // 2 x 16KB double buffer

  const int tid = threadIdx.x;
  const int bx = blockIdx.x & 7, by = blockIdx.x >> 3;
  const int split = blockIdx.y;
  const int x = bx*16 + (tid & 15);
  const int y = by*16 + (tid >> 4);
  const int p = y * IMGS + x;
  const float px = (2.0f*(float)x - 127.0f) * (1.0f/128.0f);
  const float py = (127.0f - 2.0f*(float)y) * (1.0f/128.0f);

  const char* gbase = (const char*)(rec + (size_t)split * FPS * RECF);
  const unsigned l0 = lds_off(&tile[0][0]);
  const unsigned l1 = lds_off(&tile[1][0]);
  const unsigned lane64 = (unsigned)tid * 64u;

  // prefetch tile 0: each thread stages 64B (4 x b128); 4 async ops / wave / tile
#pragma unroll
  for (int k = 0; k < 4; ++k)
    async_b128(l0 + lane64 + 16u*k, gbase, lane64 + 16u*k);

  PxState st;
  st.m = -1e30f; st.s = 0.0f; st.cr = 0.0f; st.cg = 0.0f; st.cb = 0.0f; st.prod = 1.0f;

  for (int t = 0; t < NTILES; ++t) {
    if (t + 1 < NTILES) {
      unsigned lb = ((t + 1) & 1) ? l1 : l0;
      unsigned go = (unsigned)(t + 1) * (TILE * RECB) + lane64;
#pragma unroll
      for (int k = 0; k < 4; ++k)
        async_b128(lb + lane64 + 16u*k, gbase, go + 16u*k);
      wait_async_le4();        // current tile complete; prefetch stays in flight
    } else {
      wait_async_le0();
    }
    __syncthreads();

    const float4* fr = (const float4*)&tile[t & 1][0];
    // 2-way unroll: interleaves two independent face chains (LDS broadcast + VALU/TRANS)
#pragma unroll 1
    for (int f = 0; f < TILE; f += 2, fr += 16) {
      raster_face(fr,     px, py, st);
      raster_face(fr + 8, px, py, st);
    }
    __syncthreads();   // buffer safe to overwrite by next prefetch
  }

  size_t o = (size_t)split * PIX + p;
  part[0*PART_STRIDE + o] = st.m;
  part[1*PART_STRIDE + o] = st.s;
  part[2*PART_STRIDE + o] = st.cr;
  part[3*PART_STRIDE + o] = st.cg;
  part[4*PART_STRIDE + o] = st.cb;
  part[5*PART_STRIDE + o] = st.prod;
}

// ---------------- kernel 3: merge face-splits (online-softmax fold) ----------------
__global__ __launch_bounds__(256) void sr_merge(
    const float* __restrict__ part, float* __restrict__ out) {
  int p = blockIdx.x * 256 + threadIdx.x;
  if (p >= PIX) return;
  // background element: m0 = BG_EPS, s0 = 1, c = 0, prod = 1
  float m = BG_EPS, s = 1.0f, cr = 0.0f, cg = 0.0f, cb = 0.0f, prod = 1.0f;
#pragma unroll
  for (int i = 0; i < NSPLIT; ++i) {
    size_t o = (size_t)i * PIX + p;
    float mi = part[0*PART_STRIDE + o];
    float si = part[1*PART_STRIDE + o];
    float ri = part[2*PART_STRIDE + o];
    float gi = part[3*PART_STRIDE + o];
    float bi = part[4*PART_STRIDE + o];
    float pi = part[5*PART_STRIDE + o];
    float nm = fmaxf(m, mi);
    float a  = __expf((m  - nm) * INV_G);
    float b  = __expf((mi - nm) * INV_G);
    s  = s*a  + si*b;
    cr = cr*a + ri*b;
    cg = cg*a + gi*b;
    cb = cb*a + bi*b;
    prod *= pi;
    m = nm;
  }
  float is = 1.0f / s;
  out[0*PIX + p] = cr * is;
  out[1*PIX + p] = cg * is;
  out[2*PIX + p] = cb * is;
  out[3*PIX + p] = 1.0f - prod;
}

// ---------------- host launcher ----------------
extern "C" void kernel_launch(void* const* d_in, const int* in_sizes, int n_in,
                              void* d_out, int out_size, void* d_ws, size_t ws_size,
                              hipStream_t stream) {
  const float* verts = (const float*)d_in[0];   // (1, 5023, 3) f32
  const int*   faces = (const int*)  d_in[1];   // (1, 10000, 3) i32
  const float* attr  = (const float*)d_in[2];   // (1, 10000, 3, 3) f32
  float* out = (float*)d_out;                   // (1, 4, 128, 128) f32

  float* rec  = (float*)d_ws;                   // 20480 * 32 floats (2.62 MB)
  float* part = rec + WS_PART_OFF;              // 6 * 10 * 16384 floats (3.93 MB)

  sr_setup <<<F2P/256, 256, 0, stream>>>(verts, faces, attr, rec);
  sr_render<<<dim3(64, NSPLIT), 256, 0, stream>>>(rec, part);
  sr_merge <<<PIX/256, 256, 0, stream>>>(part, out);
}